// DepthSensitiveLoss_60078002536946
// MI455X (gfx1250) — compile-verified
//
#include <hip/hip_runtime.h>
#include <hip/hip_bf16.h>

#define ALPHA_C 0.5f
#define EPS_C   1e-6f
#define N_COLS  4096
#define S1_THREADS 256
#define PER_T  (N_COLS / S1_THREADS)   // 16 contiguous elements per thread

typedef float vf4 __attribute__((ext_vector_type(4)));   // clang vector: OK for NT builtins

static __device__ __forceinline__ int imax(int a, int b) { return a > b ? a : b; }

// ---------------- Stage 1: one workgroup per row ----------------
// Streams 48KB/row (3 arrays x 16KB) with nontemporal b128 loads,
// computes sum(w*bce) and max run of correct predictions per row.
__global__ __launch_bounds__(S1_THREADS) void dsl_stage1(
    const float* __restrict__ yp, const float* __restrict__ yt,
    const float* __restrict__ dw,
    float* __restrict__ row_bce, float* __restrict__ row_streak)
{
    const int row = blockIdx.x;
    const int t   = threadIdx.x;
    const size_t base = (size_t)row * N_COLS + (size_t)t * PER_T;

    float p[PER_T], tv[PER_T], wv[PER_T];
    const vf4* p4 = reinterpret_cast<const vf4*>(yp + base);
    const vf4* t4 = reinterpret_cast<const vf4*>(yt + base);
    const vf4* w4 = reinterpret_cast<const vf4*>(dw + base);
#pragma unroll
    for (int i = 0; i < PER_T / 4; ++i) {
        vf4 a = __builtin_nontemporal_load(p4 + i);   // global_load_b128 th:NT
        vf4 b = __builtin_nontemporal_load(t4 + i);
        vf4 c = __builtin_nontemporal_load(w4 + i);
#pragma unroll
        for (int k = 0; k < 4; ++k) {
            p [4*i+k] = a[k];
            tv[4*i+k] = b[k];
            wv[4*i+k] = c[k];
        }
    }

    // per-thread sequential scan over its 16-element segment
    float bsum = 0.0f;
    int cur = 0, best = 0, pref = 0;
    bool allc = true;
#pragma unroll
    for (int j = 0; j < PER_T; ++j) {
        const float pj = p[j], tj = tv[j];
        const bool  c  = (pj > 0.5f) == (tj > 0.5f);
        const float bce = -(tj * __logf(pj + EPS_C) +
                            (1.0f - tj) * __logf(1.0f - pj + EPS_C));
        bsum = fmaf(wv[j], bce, bsum);
        if (c) { ++cur; best = imax(best, cur); if (allc) ++pref; }
        else   { cur = 0; allc = false; }
    }
    const int suf = cur;

    // ordered LDS tree reduction: (pref,best,suf) combine is associative
    __shared__ float sB   [S1_THREADS];
    __shared__ int   sPref[S1_THREADS];
    __shared__ int   sBest[S1_THREADS];
    __shared__ int   sSuf [S1_THREADS];
    sB[t] = bsum; sPref[t] = pref; sBest[t] = best; sSuf[t] = suf;
    __syncthreads();

    int len = PER_T;                       // segment length at current level
    for (int stride = 1; stride < S1_THREADS; stride <<= 1) {
        if ((t & (2 * stride - 1)) == 0) {
            const int o = t + stride;
            const int pf1 = sPref[t], bst1 = sBest[t], sf1 = sSuf[t];
            const int pf2 = sPref[o], bst2 = sBest[o], sf2 = sSuf[o];
            sBest[t] = imax(imax(bst1, bst2), sf1 + pf2);
            sPref[t] = (pf1 == len) ? len + pf2 : pf1;
            sSuf [t] = (sf2 == len) ? len + sf1 : sf2;
            sB[t]   += sB[o];
        }
        len <<= 1;
        __syncthreads();
    }

    if (t == 0) {
        row_bce[row]    = sB[0];
        row_streak[row] = (float)sBest[0];
    }
}

// ---------------- Stage 2: one wave reduces the 8192 row partials ----------------
// Uses V_WMMA_F32_16X16X4_F32 (full f32 precision) as the cross-lane summation
// engine: A = lane partials (16x4, zero-padded), B = ones -> D rows hold
// partial[m] + partial[m+16]; lane-local sum of D's 8 VGPRs + shfl_xor(16)
// yields the full 32-lane sum in every lane.
__global__ __launch_bounds__(32) void dsl_stage2(
    const float* __restrict__ row_bce, const float* __restrict__ row_streak,
    float* __restrict__ out, int rows)
{
    const int lane = threadIdx.x;
    float s1 = 0.0f, s2 = 0.0f;
    for (int i = lane; i < rows; i += 32) {
        s1 += row_bce[i];
        s2 += row_streak[i];
    }

    float tot1, tot2;
#if __has_builtin(__builtin_amdgcn_wmma_f32_16x16x4_f32)
    typedef float v2f __attribute__((ext_vector_type(2)));
    typedef float v8f __attribute__((ext_vector_type(8)));
    v2f a1; a1[0] = s1;   a1[1] = 0.0f;   // lanes 0-15: K0; lanes 16-31: K2
    v2f a2; a2[0] = s2;   a2[1] = 0.0f;
    v2f bo; bo[0] = 1.0f; bo[1] = 1.0f;   // B = all ones (layout-invariant)
    v8f cz = {};
    v8f d1 = __builtin_amdgcn_wmma_f32_16x16x4_f32(
        false, a1, false, bo, (short)0, cz, false, false);
    v8f d2 = __builtin_amdgcn_wmma_f32_16x16x4_f32(
        false, a2, false, bo, (short)0, cz, false, false);
    float r1 = 0.0f, r2 = 0.0f;
#pragma unroll
    for (int k = 0; k < 8; ++k) { r1 += d1[k]; r2 += d2[k]; }
    tot1 = r1 + __shfl_xor(r1, 16, 32);
    tot2 = r2 + __shfl_xor(r2, 16, 32);
#else
    for (int off = 16; off > 0; off >>= 1) {
        s1 += __shfl_xor(s1, off, 32);
        s2 += __shfl_xor(s2, off, 32);
    }
    tot1 = s1; tot2 = s2;
#endif

    if (lane == 0) {
        const float inv = 1.0f / ((float)rows * (float)N_COLS);
        // out = ALPHA * S1/(B*N) + (1-ALPHA) * (1 - S2/(B*N))
        out[0] = ALPHA_C * (tot1 * inv) + (1.0f - ALPHA_C) * (1.0f - tot2 * inv);
    }
}

extern "C" void kernel_launch(void* const* d_in, const int* in_sizes, int n_in,
                              void* d_out, int out_size, void* d_ws, size_t ws_size,
                              hipStream_t stream)
{
    const float* yp = (const float*)d_in[0];   // y_pred [B,N] f32
    const float* yt = (const float*)d_in[1];   // y_true [B,N] f32
    const float* dw = (const float*)d_in[2];   // depth_weights [B,N] f32
    float* out = (float*)d_out;

    const int total = in_sizes[0];
    const int rows  = total / N_COLS;          // B = 8192 for the reference shape

    float* ws         = (float*)d_ws;
    float* row_bce    = ws;                    // rows floats
    float* row_streak = ws + rows;             // rows floats

    dsl_stage1<<<rows, S1_THREADS, 0, stream>>>(yp, yt, dw, row_bce, row_streak);
    dsl_stage2<<<1, 32, 0, stream>>>(row_bce, row_streak, out, rows);
}